// GCN_9560597201561
// MI455X (gfx1250) — compile-verified
//
#include <hip/hip_runtime.h>
#include <cstdint>

#define F_IN  128
#define F_HID 128
#define F_OUT 64

typedef __attribute__((ext_vector_type(2))) float v2f;
typedef __attribute__((ext_vector_type(8))) float v8f;

__device__ __forceinline__ void atomic_add_f32(float* p, float v) {
  unsafeAtomicAdd(p, v);   // -> global_atomic_add_f32 on gfx1250
}

// ---------------- init kernels ----------------
__global__ void k_fill_const(float* __restrict__ p, float v, long long n) {
  long long i = (long long)blockIdx.x * blockDim.x + threadIdx.x;
  if (i < n) p[i] = v;
}

// out[i*64 + j] = b[j]  (bias pre-seeded so layer-2 scatter lands on top of it)
__global__ void k_fill_bias64(float* __restrict__ out, const float* __restrict__ b,
                              long long total) {
  long long i = (long long)blockIdx.x * blockDim.x + threadIdx.x;
  if (i < total) out[i] = b[i & 63];
}

// ---------------- degree / normalization ----------------
__global__ void k_deg(const int* __restrict__ col, const float* __restrict__ ew,
                      float* __restrict__ deg, int E) {
  int e = blockIdx.x * blockDim.x + threadIdx.x;
  if (e < E) atomic_add_f32(&deg[col[e]], ew[e]);
}

__global__ void k_rsqrt(float* __restrict__ deg, int n) {
  int i = blockIdx.x * blockDim.x + threadIdx.x;
  if (i < n) {
    float d = deg[i];
    deg[i] = (d > 0.f) ? rsqrtf(d) : 0.f;
  }
}

__global__ void k_norm(const int* __restrict__ row, const int* __restrict__ col,
                       const float* __restrict__ ew, const float* __restrict__ dis,
                       float* __restrict__ nrm, int E) {
  int e = blockIdx.x * blockDim.x + threadIdx.x;
  if (e < E) nrm[e] = dis[row[e]] * ew[e] * dis[col[e]];
}

// ---------------- WMMA GEMM: H[n,FOUT] = A[n,128] @ W[128,FOUT] ----------------
// One wave32 computes one 16x16 output tile; K=128 in 32 steps of 16x16x4 fp32 WMMA.
// A frag (16x4 f32, 2 VGPRs):  lanes 0-15 hold M=lane,K={k,k+1}; lanes 16-31 M=lane-16,K={k+2,k+3}
// B frag (4x16 f32, 2 VGPRs):  lanes 0-15 hold N=lane,K={k,k+1}; lanes 16-31 N=lane-16,K={k+2,k+3}
// C/D  (16x16 f32, 8 VGPRs):   VGPR v: lanes 0-15 -> (M=v, N=lane); lanes 16-31 -> (M=v+8, N=lane-16)
template <int FOUT>
__global__ void __launch_bounds__(256)
k_gemm_wmma(const float* __restrict__ A, const float* __restrict__ W,
            float* __restrict__ H, int n) {
  __shared__ float sW[F_IN * FOUT];     // 64KB (FOUT=128) / 32KB (FOUT=64) of 320KB WGP LDS
  for (int i = threadIdx.x; i < F_IN * FOUT; i += 256) sW[i] = W[i];
  __syncthreads();

  const int lane = threadIdx.x & 31;
  const int wave = threadIdx.x >> 5;
  const int tilesN = FOUT / 16;
  const long long numTiles = (long long)(n / 16) * tilesN;   // n % 16 == 0 here (50000)
  const long long tile = (long long)blockIdx.x * 8 + wave;
  if (tile >= numTiles) return;        // wave-uniform: EXEC stays all-1s for WMMA

  const int tm = (int)(tile / tilesN);
  const int tn = (int)(tile % tilesN);
  const int mrow = tm * 16 + (lane & 15);
  const int ncol = tn * 16 + (lane & 15);
  const int koff = (lane >> 4) * 2;    // 0 for lanes 0-15, 2 for lanes 16-31

  const float* __restrict__ arow = A + (long long)mrow * F_IN;
  const float* __restrict__ wb = sW + ncol;

  v8f c = {};
#pragma unroll
  for (int k = 0; k < F_IN; k += 4) {
    const int ka = k + koff;
    v2f a, b;
    a.x = arow[ka];                    // 64-bit contiguous global load
    a.y = arow[ka + 1];
    b.x = wb[ka * FOUT];               // LDS broadcast loads
    b.y = wb[(ka + 1) * FOUT];
    c = __builtin_amdgcn_wmma_f32_16x16x4_f32(
        /*neg_a=*/false, a, /*neg_b=*/false, b,
        /*c_mod=*/(short)0, c, /*reuse_a=*/false, /*reuse_b=*/false);
  }

  const int rbase = tm * 16 + ((lane >> 4) << 3);
#pragma unroll
  for (int v = 0; v < 8; ++v)
    H[(long long)(rbase + v) * FOUT + ncol] = c[v];
}

// ---------------- edge scatter: out[col] += H[row] * norm ----------------
// One wave per edge; F/32 floats per lane; vector gather + fp32 HW atomics (L2-resident).
template <int F>
__global__ void __launch_bounds__(256)
k_scatter(const float* __restrict__ H, const int* __restrict__ row,
          const int* __restrict__ col, const float* __restrict__ nrm,
          float* __restrict__ out, int E) {
  const int lane = threadIdx.x & 31;
  const long long e = (long long)blockIdx.x * 8 + (threadIdx.x >> 5);
  if (e >= E) return;
  const int r = row[e];
  const int cgt = col[e];
  const float w = nrm[e];
  constexpr int PER = F / 32;          // 4 (F=128) or 2 (F=64)
  const float* src = H + (long long)r * F + lane * PER;
  float* dst = out + (long long)cgt * F + lane * PER;
  if constexpr (PER == 4) {
    float4 v = *(const float4*)src;    // global_load_b128
    atomic_add_f32(dst + 0, v.x * w);
    atomic_add_f32(dst + 1, v.y * w);
    atomic_add_f32(dst + 2, v.z * w);
    atomic_add_f32(dst + 3, v.w * w);
  } else {
    float2 v = *(const float2*)src;    // global_load_b64
    atomic_add_f32(dst + 0, v.x * w);
    atomic_add_f32(dst + 1, v.y * w);
  }
}

// layer-1 epilogue: a1 = relu(agg + h*dis^2 (self-loop) + b1), in place into agg
__global__ void k_combine1(float* __restrict__ agg, const float* __restrict__ h,
                           const float* __restrict__ dis, const float* __restrict__ b1,
                           long long total) {
  long long i = (long long)blockIdx.x * blockDim.x + threadIdx.x;
  if (i >= total) return;
  const int node = (int)(i >> 7);
  const int j = (int)(i & 127);
  const float d = dis[node];
  float v = agg[i] + h[i] * d * d + b1[j];
  agg[i] = v > 0.f ? v : 0.f;
}

// layer-2 epilogue: out += h2*dis^2 (self-loop); bias was pre-seeded in init
__global__ void k_combine2(float* __restrict__ out, const float* __restrict__ h2,
                           const float* __restrict__ dis, long long total) {
  long long i = (long long)blockIdx.x * blockDim.x + threadIdx.x;
  if (i >= total) return;
  const int node = (int)(i >> 6);
  const float d = dis[node];
  out[i] += h2[i] * d * d;
}

static inline int nblk(long long n, int t) { return (int)((n + t - 1) / t); }

extern "C" void kernel_launch(void* const* d_in, const int* in_sizes, int n_in,
                              void* d_out, int out_size, void* d_ws, size_t ws_size,
                              hipStream_t stream) {
  const float* x  = (const float*)d_in[0];
  const int*   ei = (const int*)d_in[1];
  const float* ew = (const float*)d_in[2];
  const float* W1 = (const float*)d_in[3];
  const float* b1 = (const float*)d_in[4];
  const float* W2 = (const float*)d_in[5];
  const float* b2 = (const float*)d_in[6];
  float* out = (float*)d_out;

  const int n = in_sizes[0] / F_IN;    // 50000
  const int E = in_sizes[1] / 2;       // 800000
  const int* row = ei;                 // edge_index[0]
  const int* col = ei + E;             // edge_index[1]

  // workspace carve-up (~55 MB)
  char* ws = (char*)d_ws;
  size_t o = 0;
  auto carve = [&](size_t bytes) {
    char* p = ws + o;
    o += (bytes + 255) & ~(size_t)255;
    return p;
  };
  float* dis = (float*)carve((size_t)n * 4);             // deg -> dis, in place
  float* nrm = (float*)carve((size_t)E * 4);
  float* h   = (float*)carve((size_t)n * F_HID * 4);     // h1, later reused as h2
  float* agg = (float*)carve((size_t)n * F_HID * 4);     // layer-1 accumulator -> a1

  const long long nF1 = (long long)n * F_HID;
  const long long nF2 = (long long)n * F_OUT;

  // init: deg starts at 1.0 (self-loop weight), accumulator zero, out pre-seeded with b2
  k_fill_const<<<nblk(n, 256), 256, 0, stream>>>(dis, 1.0f, n);
  k_fill_const<<<nblk(nF1, 256), 256, 0, stream>>>(agg, 0.0f, nF1);
  k_fill_bias64<<<nblk(nF2, 256), 256, 0, stream>>>(out, b2, nF2);

  // normalization
  k_deg<<<nblk(E, 256), 256, 0, stream>>>(col, ew, dis, E);
  k_rsqrt<<<nblk(n, 256), 256, 0, stream>>>(dis, n);
  k_norm<<<nblk(E, 256), 256, 0, stream>>>(row, col, ew, dis, nrm, E);

  // ---- layer 1 ----
  {
    const long long tiles = (long long)(n / 16) * (F_HID / 16);
    k_gemm_wmma<F_HID><<<nblk(tiles, 8), 256, 0, stream>>>(x, W1, h, n);
  }
  k_scatter<F_HID><<<nblk((long long)E, 8), 256, 0, stream>>>(h, row, col, nrm, agg, E);
  k_combine1<<<nblk(nF1, 256), 256, 0, stream>>>(agg, h, dis, b1, nF1);

  // ---- layer 2 ---- (h buffer reused for h2)
  {
    const long long tiles = (long long)(n / 16) * (F_OUT / 16);
    k_gemm_wmma<F_OUT><<<nblk(tiles, 8), 256, 0, stream>>>(agg, W2, h, n);
  }
  k_scatter<F_OUT><<<nblk((long long)E, 8), 256, 0, stream>>>(h, row, col, nrm, out, E);
  k_combine2<<<nblk(nF2, 256), 256, 0, stream>>>(out, h, dis, nF2);
}